// TripletLoss_6004364280443
// MI455X (gfx1250) — compile-verified
//
#include <hip/hip_runtime.h>
#include <hip/hip_bf16.h>

typedef float v2f __attribute__((ext_vector_type(2)));
typedef float v4f __attribute__((ext_vector_type(4)));
typedef float v8f __attribute__((ext_vector_type(8)));

#define MARGIN 1.0f

// ---------------------------------------------------------------------------
// Kernel 1: per-row squared norms  sq[i] = sum_d emb[i,d]^2   (float4 loads)
// ---------------------------------------------------------------------------
__global__ void k_sqnorm(const float* __restrict__ emb, float* __restrict__ sq,
                         int N, int D) {
    int i = blockIdx.x * blockDim.x + threadIdx.x;
    if (i < N) {
        const v4f* row = (const v4f*)(emb + (long)i * D);
        float s = 0.0f;
        for (int d = 0; d < (D >> 2); ++d) {
            v4f v = row[d];
            s = fmaf(v.x, v.x, s);
            s = fmaf(v.y, v.y, s);
            s = fmaf(v.z, v.z, s);
            s = fmaf(v.w, v.w, s);
        }
        sq[i] = s;
    }
}

// ---------------------------------------------------------------------------
// Kernel 2: pdist[i,j] = max(sq_i + sq_j - 2 * <e_i, e_j>, 0), diag zeroed.
// One wave32 per 16x16 tile; Gram via V_WMMA_F32_16X16X4_F32, D/4 k-steps.
// A lane L: row M = L%16, K-pair sel=L/16 -> 2 consecutive K values (float2).
// B is emb.T tile => same addressing with the column-block row base.
// ---------------------------------------------------------------------------
__global__ __launch_bounds__(256) void k_pdist_wmma(
    const float* __restrict__ emb, const float* __restrict__ sq,
    float* __restrict__ pdist, int N, int D) {
    const int lane = threadIdx.x & 31;
    const int wave = threadIdx.x >> 5;
    const int tilesPerRow = N >> 4;
    const int nTiles = tilesPerRow * tilesPerRow;
    const int tile = blockIdx.x * 8 + wave;
    if (tile >= nTiles) return;            // whole-wave exit: EXEC all-ones inside

    const int ti = tile / tilesPerRow;
    const int tj = tile - ti * tilesPerRow;
    const int i0 = ti << 4, j0 = tj << 4;
    const int m   = lane & 15;
    const int sel = lane >> 4;             // 0: K={0,1}, 1: K={2,3}

    const float* aBase = emb + (long)(i0 + m) * D + 2 * sel;
    const float* bBase = emb + (long)(j0 + m) * D + 2 * sel;

    v8f c = {};
    #pragma unroll 8
    for (int kk = 0; kk < D; kk += 4) {
        v2f a = *(const v2f*)(aBase + kk);   // global_load_b64, 8B aligned
        v2f b = *(const v2f*)(bBase + kk);
        c = __builtin_amdgcn_wmma_f32_16x16x4_f32(
                /*neg_a=*/false, a, /*neg_b=*/false, b,
                /*c_mod=*/(short)0, c, /*reuse_a=*/false, /*reuse_b=*/false);
    }

    // This lane writes M = r + 8*sel, N-col = m. Row norms needed: the 8
    // consecutive values sq[i0 + 8*sel .. +7] -> two b128 loads (32B aligned).
    const float sqn = sq[j0 + m];
    const v4f* sqv = (const v4f*)(sq + i0 + (sel << 3));
    v4f sqLo = sqv[0], sqHi = sqv[1];
    float sqi[8] = {sqLo.x, sqLo.y, sqLo.z, sqLo.w, sqHi.x, sqHi.y, sqHi.z, sqHi.w};

    #pragma unroll
    for (int r = 0; r < 8; ++r) {
        const int M  = r + (sel << 3);       // C/D layout: lanes>=16 hold M+8
        const int gi = i0 + M;
        const int gj = j0 + m;
        float v = sqi[r] + sqn - 2.0f * c[r];
        v = fmaxf(v, 0.0f);
        if (gi == gj) v = 0.0f;
        pdist[(long)gi * N + gj] = v;
    }
}

// ---------------------------------------------------------------------------
// Kernel 3: semi-hard mining, one row i per block. Row + label-fused masked
// row live in LDS; inner k-scan is ONE LDS broadcast serving BOTH columns a
// thread owns (j, j+256). mskD[k] = -INF for same-label k folds the label
// test into the single compare (dk > dij). Per-row partial sums out.
// ---------------------------------------------------------------------------
__global__ __launch_bounds__(256) void k_mine(
    const float* __restrict__ pdist, const int* __restrict__ label,
    float* __restrict__ partials, int N) {
    extern __shared__ char smem[];
    float* rowD = (float*)smem;              // N floats: pdist row i
    float* mskD = rowD + N;                  // N floats: rowD masked by label
    int*   lab  = (int*)(mskD + N);          // N ints
    float* red  = (float*)(lab + N);         // 256 floats

    const int i = blockIdx.x;
    const int t = threadIdx.x;

    __builtin_prefetch(pdist + (long)i * N + t, 0, 1);   // global_prefetch_b8

    for (int j = t; j < N; j += 256) {
        rowD[j] = pdist[(long)i * N + j];
        lab[j]  = label[j];
    }
    __syncthreads();
    const int li = lab[i];

    // ---- masked row + row max / min ----
    float mx = -INFINITY, mn = INFINITY;
    for (int j = t; j < N; j += 256) {
        float v = rowD[j];
        mskD[j] = (lab[j] != li) ? v : -INFINITY;
        mx = fmaxf(mx, v);
        mn = fminf(mn, v);
    }
    red[t] = mx; __syncthreads();
    for (int s = 128; s > 0; s >>= 1) {
        if (t < s) red[t] = fmaxf(red[t], red[t + s]);
        __syncthreads();
    }
    const float rowmax = red[0]; __syncthreads();
    red[t] = mn; __syncthreads();
    for (int s = 128; s > 0; s >>= 1) {
        if (t < s) red[t] = fminf(red[t], red[t + s]);
        __syncthreads();
    }
    const float rowmin = red[0]; __syncthreads();

    // ---- negatives_inside = max_j (D[i,j]-rowmin)*adjnot + rowmin ----
    float ni = -INFINITY;
    for (int j = t; j < N; j += 256) {
        float adjnot = (lab[j] != li) ? 1.0f : 0.0f;
        ni = fmaxf(ni, (rowD[j] - rowmin) * adjnot);
    }
    red[t] = ni; __syncthreads();
    for (int s = 128; s > 0; s >>= 1) {
        if (t < s) red[t] = fmaxf(red[t], red[t + s]);
        __syncthreads();
    }
    const float neg_inside = red[0] + rowmin; __syncthreads();

    // ---- dual-column semi-hard scan + loss accumulation ----
    float lsum = 0.0f, lcnt = 0.0f;
    for (int jb = 0; jb < N; jb += 512) {
        const int j0 = jb + t;
        const int j1 = jb + t + 256;
        const float d0 = (j0 < N) ? rowD[j0] : INFINITY;
        const float d1 = (j1 < N) ? rowD[j1] : INFINITY;
        float mv0 = INFINITY, mv1 = INFINITY;
        int a0 = 0, a1 = 0;
        #pragma unroll 4
        for (int k = 0; k < N; ++k) {
            const float dk = mskD[k];        // uniform addr -> LDS broadcast
            const float v  = dk - rowmax;
            const bool m0 = dk > d0;         // false for same-label (dk=-INF)
            const bool m1 = dk > d1;
            mv0 = fminf(mv0, m0 ? v : 0.0f); a0 |= (int)m0;
            mv1 = fminf(mv1, m1 ? v : 0.0f); a1 |= (int)m1;
        }
        if (j0 < N) {
            const float shn  = a0 ? (mv0 + rowmax) : neg_inside;
            const float mpos = ((lab[j0] == li) && (j0 != i)) ? 1.0f : 0.0f;
            lsum += fmaxf((MARGIN + d0 - shn) * mpos, 0.0f);
            lcnt += mpos;
        }
        if (j1 < N) {
            const float shn  = a1 ? (mv1 + rowmax) : neg_inside;
            const float mpos = ((lab[j1] == li) && (j1 != i)) ? 1.0f : 0.0f;
            lsum += fmaxf((MARGIN + d1 - shn) * mpos, 0.0f);
            lcnt += mpos;
        }
    }
    red[t] = lsum; __syncthreads();
    for (int s = 128; s > 0; s >>= 1) {
        if (t < s) red[t] += red[t + s];
        __syncthreads();
    }
    if (t == 0) partials[i] = red[0];
    __syncthreads();
    red[t] = lcnt; __syncthreads();
    for (int s = 128; s > 0; s >>= 1) {
        if (t < s) red[t] += red[t + s];
        __syncthreads();
    }
    if (t == 0) partials[N + i] = red[0];
}

// ---------------------------------------------------------------------------
// Kernel 4: deterministic fixed-order final reduction: loss_sum / num_pos
// ---------------------------------------------------------------------------
__global__ void k_final(const float* __restrict__ partials,
                        float* __restrict__ out, int N) {
    if (threadIdx.x == 0 && blockIdx.x == 0) {
        float s = 0.0f, c = 0.0f;
        for (int b = 0; b < N; ++b) { s += partials[b]; c += partials[N + b]; }
        out[0] = s / c;
    }
}

extern "C" void kernel_launch(void* const* d_in, const int* in_sizes, int n_in,
                              void* d_out, int out_size, void* d_ws, size_t ws_size,
                              hipStream_t stream) {
    const float* emb   = (const float*)d_in[0];
    const int*   label = (const int*)d_in[1];
    const int N = in_sizes[1];
    const int D = in_sizes[0] / N;

    float* pdist    = (float*)d_ws;                  // N*N floats
    float* sq       = pdist + (size_t)N * N;         // N floats
    float* partials = sq + N;                        // 2*N floats
    float* out      = (float*)d_out;

    k_sqnorm<<<(N + 255) / 256, 256, 0, stream>>>(emb, sq, N, D);

    const int tiles = (N / 16) * (N / 16);
    k_pdist_wmma<<<(tiles + 7) / 8, 256, 0, stream>>>(emb, sq, pdist, N, D);

    const size_t shmem = (size_t)N * sizeof(float) * 2     // rowD + mskD
                       + (size_t)N * sizeof(int)           // lab
                       + 256 * sizeof(float);              // red
    k_mine<<<N, 256, shmem, stream>>>(pdist, label, partials, N);

    k_final<<<1, 1, 0, stream>>>(partials, out, N);
}